// slater_wf_momentum_21973052686657
// MI455X (gfx1250) — compile-verified
//
#include <hip/hip_runtime.h>
#include <math.h>

typedef __attribute__((ext_vector_type(2))) float v2f;
typedef __attribute__((ext_vector_type(8))) float v8f;

#define BB    32
#define NORB  2
#define LX    12
#define LY    12
#define LSITE 144
#define NEL   36
#define NLOC  2
#define KTOT  288      // LSITE*NLOC
#define NOUT  5184     // LSITE*NEL
#define MTOT  64       // BB*NORB
#define MPITCH 37      // 36 + 1 pad (odd -> conflict free)
#define UVAL  4.0f
#define N0VAL 0.5f

// ---------------------------------------------------------------------------
// Kernel 1: SM_full = x @ W + b   (fp32 WMMA 16x16x4, M=64, K=288, N=5184)
// grid (81, 4), block 128 (4 waves); each wave owns one 16-wide N tile.
// ---------------------------------------------------------------------------
__global__ void __launch_bounds__(128)
gemm_wmma_f32(const float* __restrict__ x, const float* __restrict__ W,
              const float* __restrict__ bias, float* __restrict__ SM_full)
{
    __shared__ float As[16 * KTOT];          // 18 KB A tile for this M-block
    const int tid   = threadIdx.x;
    const int mbase = blockIdx.y * 16;

    for (int i = tid; i < 16 * KTOT; i += 128) {
        int m = i / KTOT, k = i % KTOT;
        As[i] = x[(size_t)(mbase + m) * KTOT + k];
    }
    __syncthreads();

    const int wave  = tid >> 5;
    const int lane  = tid & 31;
    const int lhalf = lane & 15;
    const bool hi   = lane >= 16;
    const int ncol  = (blockIdx.x * 4 + wave) * 16 + lhalf;

    v8f c = {0.f, 0.f, 0.f, 0.f, 0.f, 0.f, 0.f, 0.f};

    for (int k = 0; k < KTOT; k += 4) {
        // A 16x4 fp32: lanes 0-15 -> K=k,k+1 ; lanes 16-31 -> K=k+2,k+3 ; row = lhalf
        const int ka = k + (hi ? 2 : 0);
        v2f a, bf;
        a[0] = As[lhalf * KTOT + ka];
        a[1] = As[lhalf * KTOT + ka + 1];
        // B 4x16 fp32: VGPR0 rows {k | k+2}, VGPR1 rows {k+1 | k+3}; col = lhalf
        const float* wp = W + (size_t)ka * NOUT + ncol;
        __builtin_prefetch(wp + 4 * NOUT, 0, 1);     // stream next K-slab (global_prefetch)
        bf[0] = wp[0];
        bf[1] = wp[NOUT];
        c = __builtin_amdgcn_wmma_f32_16x16x4_f32(false, a, false, bf,
                                                  (short)0, c, false, false);
    }

    // D layout: VGPR v -> M = v (+8 for hi lanes), N = lhalf
#pragma unroll
    for (int v = 0; v < 8; ++v) {
        int m = mbase + v + (hi ? 8 : 0);
        SM_full[(size_t)m * NOUT + ncol] = c[v] + bias[ncol];
    }
}

// ---------------------------------------------------------------------------
// Shared 36x36 LU (partial pivoting) on a mean-normalized matrix in LDS.
// 64 threads (2 waves). Returns log|det| + 36*log(mean) and sign in sh2[0..1].
// ---------------------------------------------------------------------------
__device__ __forceinline__ void lu36(float* M, float* rv, int* ri, float* sh2, int tid,
                                     float* out_ld, float* out_sg)
{
    // mean(|M|)
    float acc = 0.f;
    for (int i = tid; i < NEL * NEL; i += 64)
        acc += fabsf(M[(i / NEL) * MPITCH + (i % NEL)]);
    rv[tid] = acc; __syncthreads();
    for (int s = 32; s > 0; s >>= 1) { if (tid < s) rv[tid] += rv[tid + s]; __syncthreads(); }
    const float mean  = rv[0] * (1.0f / (NEL * NEL));
    __syncthreads();
    const float rmean = 1.0f / mean;
    for (int i = tid; i < NEL * NEL; i += 64)
        M[(i / NEL) * MPITCH + (i % NEL)] *= rmean;
    if (tid == 0) { sh2[0] = 0.f; sh2[1] = 1.f; }
    __syncthreads();

    for (int k = 0; k < NEL; ++k) {
        // arg-max pivot in column k, rows k..35
        float v = -1.f;
        if (tid >= k && tid < NEL) v = fabsf(M[tid * MPITCH + k]);
        rv[tid] = v; ri[tid] = tid; __syncthreads();
        for (int s = 32; s > 0; s >>= 1) {
            if (tid < s && rv[tid + s] > rv[tid]) { rv[tid] = rv[tid + s]; ri[tid] = ri[tid + s]; }
            __syncthreads();
        }
        const int p = ri[0]; __syncthreads();
        if (p != k && tid < NEL) {
            float t = M[k * MPITCH + tid];
            M[k * MPITCH + tid] = M[p * MPITCH + tid];
            M[p * MPITCH + tid] = t;
        }
        __syncthreads();
        const float piv = M[k * MPITCH + k];
        if (tid == 0) {
            sh2[0] += logf(fabsf(piv) + 1e-30f);
            float sg = sh2[1];
            if (piv < 0.f) sg = -sg;
            if (p != k)    sg = -sg;
            sh2[1] = sg;
        }
        const float rpiv = 1.0f / piv;
        const int i = k + 1 + tid;
        if (i < NEL) {
            const float f = M[i * MPITCH + k] * rpiv;
            for (int j = k + 1; j < NEL; ++j)
                M[i * MPITCH + j] -= f * M[k * MPITCH + j];
        }
        __syncthreads();
    }
    *out_ld = sh2[0] + (float)NEL * logf(mean);
    *out_sg = sh2[1];
}

// ---------------------------------------------------------------------------
// Kernel 2: per-(b,orb) gather + logdet/sign.  grid 64, block 64.
// ---------------------------------------------------------------------------
__global__ void __launch_bounds__(64)
logdet_kernel(const float* __restrict__ SM_full, const int* __restrict__ R,
              float* __restrict__ ld_orb, float* __restrict__ sg_orb)
{
    __shared__ float M[NEL * MPITCH];
    __shared__ float rv[64]; __shared__ int ri[64]; __shared__ float sh2[2];
    const int bo  = blockIdx.x;
    const int tid = threadIdx.x;
    const int* Rp = R + bo * NEL;
    const float* SMb = SM_full + (size_t)bo * NOUT;

    for (int i = tid; i < NEL * NEL; i += 64) {
        int row = i / NEL, col = i % NEL;
        M[row * MPITCH + col] = SMb[Rp[row] * NEL + col];
    }
    __syncthreads();
    float ld, sg;
    lu36(M, rv, ri, sh2, tid, &ld, &sg);
    if (tid == 0) { ld_orb[bo] = ld; sg_orb[bo] = sg; }
}

// ---------------------------------------------------------------------------
// Kernel 3: combine per-orbital results -> logdet[b], sign[b], im_hx[b]=0
// ---------------------------------------------------------------------------
__global__ void combine_kernel(const float* __restrict__ ld_orb,
                               const float* __restrict__ sg_orb,
                               float* __restrict__ out)
{
    int b = threadIdx.x;
    if (b < BB) {
        out[b]          = ld_orb[2 * b] + ld_orb[2 * b + 1];
        out[BB + b]     = sg_orb[2 * b] * sg_orb[2 * b + 1];
        out[3 * BB + b] = 0.0f;                       // im_hx
    }
}

// ---------------------------------------------------------------------------
// Kernel 4: Hubbard-U density term -> initializes re_hx[b].  grid 32, block 64.
// ---------------------------------------------------------------------------
__global__ void __launch_bounds__(64)
fill_kernel(const float* __restrict__ x, float* __restrict__ out)
{
    __shared__ float rv[64];
    const int b = blockIdx.x, tid = threadIdx.x;
    float acc = 0.f;
    for (int l = tid; l < LSITE; l += 64) {
        float f = x[((size_t)(b * NORB + 0) * LSITE + l) * NLOC + 1]
                + x[((size_t)(b * NORB + 1) * LSITE + l) * NLOC + 1] - N0VAL;
        acc += f * f;
    }
    rv[tid] = acc; __syncthreads();
    for (int s = 32; s > 0; s >>= 1) { if (tid < s) rv[tid] += rv[tid + s]; __syncthreads(); }
    if (tid == 0) out[2 * BB + b] = rv[0] * 0.5f * UVAL;
}

// ---------------------------------------------------------------------------
// Kernel 5: hop candidates. grid = 32*2*36*4 = 9216 blocks, block 64.
// SM_full_new = SM_full + (W[2r] - W[2r+1] - W[2r2] + W[2r2+1]) gathered rows.
// ---------------------------------------------------------------------------
__global__ void __launch_bounds__(64)
hop_kernel(const float* __restrict__ SM_full, const float* __restrict__ W,
           const float* __restrict__ x, const int* __restrict__ R,
           const float* __restrict__ ld_orb, const float* __restrict__ sg_orb,
           float* __restrict__ out)
{
    __shared__ float M[NEL * MPITCH];
    __shared__ float rv[64]; __shared__ int ri[64]; __shared__ float sh2[2];

    int idx = blockIdx.x;
    const int d = idx & 3;  idx >>= 2;
    const int e = idx % NEL; idx /= NEL;
    const int o = idx & 1;
    const int b = idx >> 1;
    const int bo = b * NORB + o;
    const int tid = threadIdx.x;

    const int dx = (d == 1 ? 1 : 0) - (d == 3 ? 1 : 0);
    const int dy = (d == 0 ? 1 : 0) - (d == 2 ? 1 : 0);

    const int* Rp = R + bo * NEL;
    const int r   = Rp[e];
    const int ix  = r / LY, iy = r % LY;
    const int r2  = ((ix + dx + LX) % LX) * LY + ((iy + dy + LY) % LY);

    // allowed iff target site empty in this orbital (uniform over block)
    if (x[((size_t)bo * LSITE + r2) * NLOC + 0] <= 0.5f) return;

    const size_t w0 = (size_t)(2 * r)      * NOUT;
    const size_t w1 = (size_t)(2 * r + 1)  * NOUT;
    const size_t w2 = (size_t)(2 * r2)     * NOUT;
    const size_t w3 = (size_t)(2 * r2 + 1) * NOUT;
    const float* SMb = SM_full + (size_t)bo * NOUT;

    for (int i = tid; i < NEL * NEL; i += 64) {
        int row = i / NEL, col = i % NEL;
        int site = (row == e) ? r2 : Rp[row];
        int cidx = site * NEL + col;
        float delta = W[w0 + cidx] - W[w1 + cidx] - W[w2 + cidx] + W[w3 + cidx];
        M[row * MPITCH + col] = SMb[cidx] + delta;
    }
    __syncthreads();

    float ld, sg;
    lu36(M, rv, ri, sh2, tid, &ld, &sg);
    if (tid == 0) {
        // unchanged orbital cancels in exp(ld_new - ld)*sg_new*sg
        float val = expf(ld - ld_orb[bo]) * sg * sg_orb[bo];   // * T (=1)
        atomicAdd(&out[2 * BB + b], val);
    }
}

// ---------------------------------------------------------------------------
extern "C" void kernel_launch(void* const* d_in, const int* in_sizes, int n_in,
                              void* d_out, int out_size, void* d_ws, size_t ws_size,
                              hipStream_t stream)
{
    const float* x    = (const float*)d_in[0];   // (32,2,144,2)
    const float* W    = (const float*)d_in[1];   // (288,5184)
    const float* bias = (const float*)d_in[2];   // (5184,)
    const int*   R    = (const int*)  d_in[3];   // (32,2,36)
    float* out = (float*)d_out;                  // [logdet|sign|re_hx|im_hx] 4x32

    float* SM_full = (float*)d_ws;               // 64*5184 floats (~1.3 MB)
    float* ld_orb  = SM_full + (size_t)MTOT * NOUT;
    float* sg_orb  = ld_orb + MTOT;

    dim3 g1(NOUT / 64, MTOT / 16);               // (81, 4)
    gemm_wmma_f32<<<g1, 128, 0, stream>>>(x, W, bias, SM_full);
    logdet_kernel<<<MTOT, 64, 0, stream>>>(SM_full, R, ld_orb, sg_orb);
    combine_kernel<<<1, 32, 0, stream>>>(ld_orb, sg_orb, out);
    fill_kernel<<<BB, 64, 0, stream>>>(x, out);
    hop_kernel<<<BB * NORB * NEL * 4, 64, 0, stream>>>(SM_full, W, x, R,
                                                       ld_orb, sg_orb, out);
}